// SPGNNLayers_45844480918206
// MI455X (gfx1250) — compile-verified
//
#include <hip/hip_runtime.h>
#include <hip/hip_bf16.h>
#include <stdint.h>

#define HID 64
#define EPSF 1e-10f

// ---- float <-> monotone uint key (for atomicMax on floats) ----
__device__ __forceinline__ unsigned fkey(float f) {
  unsigned b = __float_as_uint(f);
  return (b & 0x80000000u) ? ~b : (b | 0x80000000u);
}
__device__ __forceinline__ float funkey(unsigned u) {
  unsigned b = (u & 0x80000000u) ? (u & 0x7fffffffu) : ~u;
  return __uint_as_float(b);
}
// Low 32 bits of a flat LDS pointer == LDS byte offset (aperture lives in [63:32]).
__device__ __forceinline__ unsigned ldsOff(const void* p) {
  return (unsigned)(unsigned long long)p;
}

// ---------------- zero fill (out_t | exp_sum | gmax are contiguous) -------------
__global__ void k_fill0(float* __restrict__ p, size_t n) {
  size_t i = (size_t)blockIdx.x * blockDim.x + threadIdx.x;
  if (i < n) p[i] = 0.0f;
}

// ---------------- x (64,N) -> xt (N,64), LDS-tiled, fully coalesced -------------
__global__ void __launch_bounds__(256) k_transpose_in(const float* __restrict__ x,
                                                      float* __restrict__ xt, int N) {
  __shared__ float tile[HID][HID + 1];
  int n0 = blockIdx.x * HID;
  int tid = threadIdx.x;
  int nl = tid & 63, q = tid >> 6;
  int n = n0 + nl;
  if (n < N) {
#pragma unroll
    for (int r = 0; r < 16; ++r) {
      int h = q * 16 + r;
      tile[h][nl] = x[(size_t)h * N + n];  // coalesced over n
    }
  }
  __syncthreads();
  int h = tid & 63;
#pragma unroll
  for (int r = 0; r < 16; ++r) {
    int nn = n0 + q * 16 + r;
    if (nn < N) xt[(size_t)nn * HID + h] = tile[h][q * 16 + r];  // coalesced over h
  }
}

// ---------------- out_t (N,64) -> out (64,N) ------------------------------------
__global__ void __launch_bounds__(256) k_transpose_out(const float* __restrict__ ot,
                                                       float* __restrict__ out, int N) {
  __shared__ float tile[HID][HID + 1];
  int n0 = blockIdx.x * HID;
  int tid = threadIdx.x;
  int h = tid & 63, q = tid >> 6;
#pragma unroll
  for (int r = 0; r < 16; ++r) {
    int n = n0 + q * 16 + r;
    if (n < N) tile[h][q * 16 + r] = ot[(size_t)n * HID + h];  // coalesced over h
  }
  __syncthreads();
  int nl = tid & 63;
  int n = n0 + nl;
  if (n < N) {
#pragma unroll
    for (int r = 0; r < 16; ++r) {
      int hh = q * 16 + r;
      out[(size_t)hh * N + n] = tile[hh][nl];  // coalesced over n
    }
  }
}

// ---------------- global max of gathered edge weights (wave32 reduce) -----------
__global__ void __launch_bounds__(256) k_max(const float* __restrict__ ew,
                                             const long long* __restrict__ e2id,
                                             long long E, unsigned* __restrict__ gmax) {
  long long i = (long long)blockIdx.x * blockDim.x + threadIdx.x;
  long long stride = (long long)gridDim.x * blockDim.x;
  float m = -3.402823466e38f;
  for (; i < E; i += stride) m = fmaxf(m, ew[e2id[i]]);
#pragma unroll
  for (int off = 16; off; off >>= 1) m = fmaxf(m, __shfl_xor(m, off, 32));  // wave32
  if ((threadIdx.x & 31) == 0) atomicMax(gmax, fkey(m));
}

// ---------------- exp + atomic segment sum over source --------------------------
__global__ void __launch_bounds__(256) k_exp(const float* __restrict__ ew,
                                             const long long* __restrict__ e2id,
                                             const long long* __restrict__ src,
                                             const unsigned* __restrict__ gmax,
                                             float* __restrict__ exp_v,
                                             float* __restrict__ exp_sum, long long E) {
  long long i = (long long)blockIdx.x * blockDim.x + threadIdx.x;
  if (i < E) {
    float gm = funkey(*gmax);
    float v = __expf(ew[e2id[i]] - gm);
    exp_v[i] = v;
    unsafeAtomicAdd(&exp_sum[(int)src[i]], v);  // global_atomic_add_f32
  }
}

// ---------------- exp_sum -> 1/(exp_sum+eps) ------------------------------------
__global__ void __launch_bounds__(256) k_inv(float* __restrict__ es, int N) {
  int i = blockIdx.x * blockDim.x + threadIdx.x;
  if (i < N) es[i] = 1.0f / (es[i] + EPSF);
}

// ---------------- main scatter: out_t[src] += coef * xt[tgt] --------------------
// 256 edges/block staged into LDS via CDNA5 async-to-LDS; 16 lanes x float4 / edge.
__global__ void __launch_bounds__(256) k_scatter(const long long* __restrict__ src,
                                                 const long long* __restrict__ tgt,
                                                 const float* __restrict__ exp_v,
                                                 const float* __restrict__ inv_sum,
                                                 const float* __restrict__ xt,
                                                 float* __restrict__ out_t, long long E) {
  __shared__ __align__(16) long long s_src[256];
  __shared__ __align__(16) long long s_tgt[256];
  __shared__ __align__(16) float s_cf[256];
  int tid = threadIdx.x;
  long long e0 = (long long)blockIdx.x * 256;
  long long remll = E - e0;
  int rem = (remll >= 256) ? 256 : (int)remll;

  if (rem == 256) {
    // Stage the edge tile with async DMA to LDS (ASYNCcnt path, gfx1250).
    if (tid < 128) {
      unsigned l = ldsOff(&s_src[2 * tid]);
      const void* g = (const void*)(src + e0 + 2 * tid);
      asm volatile("global_load_async_to_lds_b128 %0, %1, off" :: "v"(l), "v"(g) : "memory");
    } else {
      int t2 = tid - 128;
      unsigned l = ldsOff(&s_tgt[2 * t2]);
      const void* g = (const void*)(tgt + e0 + 2 * t2);
      asm volatile("global_load_async_to_lds_b128 %0, %1, off" :: "v"(l), "v"(g) : "memory");
    }
    if (tid < 64) {
      unsigned l = ldsOff(&s_cf[4 * tid]);
      const void* g = (const void*)(exp_v + e0 + 4 * tid);
      asm volatile("global_load_async_to_lds_b128 %0, %1, off" :: "v"(l), "v"(g) : "memory");
    }
    asm volatile("s_wait_asynccnt 0" ::: "memory");
  } else {
    if (tid < rem) {
      s_src[tid] = src[e0 + tid];
      s_tgt[tid] = tgt[e0 + tid];
      s_cf[tid]  = exp_v[e0 + tid];
    }
  }
  __syncthreads();

  // One L2-resident gather per edge; fold softmax denominator into the coef.
  if (tid < rem) s_cf[tid] *= inv_sum[(int)s_src[tid]];
  __syncthreads();

  int lane4 = tid & 15;   // which float4 of the 64 hidden dims
  int eslot = tid >> 4;   // 16 edges in flight per block pass
  for (int i = eslot; i < rem; i += 16) {
    int s = (int)s_src[i];
    int t = (int)s_tgt[i];
    float c = s_cf[i];
    float4 v = ((const float4*)(xt + (size_t)t * HID))[lane4];  // 256B contiguous gather
    float* op = out_t + (size_t)s * HID + (lane4 << 2);
    unsafeAtomicAdd(op + 0, c * v.x);
    unsafeAtomicAdd(op + 1, c * v.y);
    unsafeAtomicAdd(op + 2, c * v.z);
    unsafeAtomicAdd(op + 3, c * v.w);
  }
}

extern "C" void kernel_launch(void* const* d_in, const int* in_sizes, int n_in,
                              void* d_out, int out_size, void* d_ws, size_t ws_size,
                              hipStream_t stream) {
  const float*     x    = (const float*)d_in[0];        // (64, N)
  const long long* el   = (const long long*)d_in[1];    // (2, E) int64
  const long long* e2id = (const long long*)d_in[2];    // (E,)   int64
  const float*     ew   = (const float*)d_in[3];        // (E,)

  const int H = HID;
  const int N = in_sizes[0] / H;
  const long long E = in_sizes[2];
  const long long* src = el;       // row 0
  const long long* tgt = el + E;   // row 1

  // Workspace layout (floats, all 16B aligned):
  //   xt[N*H] | exp_v[E] | out_t[N*H] | exp_sum[N] | gmax(1)
  size_t NH = (size_t)N * H;
  float* xt      = (float*)d_ws;
  float* exp_v   = xt + NH;
  float* out_t   = exp_v + (size_t)E;
  float* exp_sum = out_t + NH;
  unsigned* gmax = (unsigned*)(exp_sum + N);

  size_t nz = NH + (size_t)N + 1;  // out_t + exp_sum + gmax are contiguous
  k_fill0<<<(unsigned)((nz + 255) / 256), 256, 0, stream>>>(out_t, nz);

  k_transpose_in<<<(N + HID - 1) / HID, 256, 0, stream>>>(x, xt, N);

  k_max<<<1024, 256, 0, stream>>>(ew, e2id, E, gmax);

  unsigned eblocks = (unsigned)((E + 255) / 256);
  k_exp<<<eblocks, 256, 0, stream>>>(ew, e2id, src, gmax, exp_v, exp_sum, E);

  k_inv<<<(N + 255) / 256, 256, 0, stream>>>(exp_sum, N);

  k_scatter<<<eblocks, 256, 0, stream>>>(src, tgt, exp_v, exp_sum, xt, out_t, E);

  k_transpose_out<<<(N + HID - 1) / HID, 256, 0, stream>>>(out_t, (float*)d_out, N);
}